// CDECF_28295244546622
// MI455X (gfx1250) — compile-verified
//
#include <hip/hip_runtime.h>
#include <hip/hip_bf16.h>
#include <math.h>

#define DD        64
#define BATCH     8192
#define NUSERS    50000
#define XCOLS     128          // [u | i] concatenated

typedef float v2f __attribute__((ext_vector_type(2)));
typedef float v8f __attribute__((ext_vector_type(8)));

// ---------------------------------------------------------------------------
// x0 = [user_emb[users], item_emb[items]]   (8192 x 128)
// ---------------------------------------------------------------------------
__global__ void k_init_x(const int* __restrict__ users,
                         const int* __restrict__ items,
                         const float* __restrict__ uemb,
                         const float* __restrict__ iemb,
                         float* __restrict__ x) {
    int idx = blockIdx.x * blockDim.x + threadIdx.x;
    if (idx >= BATCH * XCOLS) return;
    int b = idx >> 7;
    int c = idx & 127;
    float v = (c < DD) ? uemb[users[b] * DD + c]
                       : iemb[items[b] * DD + (c - DD)];
    x[idx] = v;
}

// ---------------------------------------------------------------------------
// zero the effect buffer (8192 x 128)
// ---------------------------------------------------------------------------
__global__ void k_zero(float* __restrict__ p, int n) {
    for (int i = blockIdx.x * blockDim.x + threadIdx.x; i < n;
         i += gridDim.x * blockDim.x)
        p[i] = 0.0f;
}

// ---------------------------------------------------------------------------
// Graph scatter: effect[src_slot] += val * x[dst_slot]  (64 cols per edge)
// Only edges with src AND dst in the active windows contribute:
//   node n active iff n < BATCH (user half, col off 0)
//             or NUSERS <= n < NUSERS+BATCH (item half, col off 64)
// Wave loads 32 edges, ballots survivors, then cooperatively processes each
// surviving edge with all 32 lanes (2 columns per lane) using f32 atomics.
// ---------------------------------------------------------------------------
__device__ __forceinline__ int remap_node(int n) {
    // returns float-element offset into x/effect (row*128 + halfOffset), or -1
    if (n < BATCH)                        return n * XCOLS;            // user
    int j = n - NUSERS;
    if (j >= 0 && j < BATCH)              return j * XCOLS + DD;       // item
    return -1;
}

__global__ void k_scatter(const int*  __restrict__ esrc,
                          const int*  __restrict__ edst,
                          const float* __restrict__ eval,
                          const float* __restrict__ x,
                          float* __restrict__ eff,
                          int nedges) {
    const int lane    = threadIdx.x & 31;
    const int gwave   = (blockIdx.x * blockDim.x + threadIdx.x) >> 5;
    const int nwaves  = (gridDim.x * blockDim.x) >> 5;
    const int stride  = nwaves * 32;

    for (int e0 = gwave * 32; e0 < nedges; e0 += stride) {
        int e = e0 + lane;
        // stream-prefetch the next chunk of the edge arrays
        if (e + stride < nedges) {
            __builtin_prefetch(&esrc[e + stride], 0, 0);
            __builtin_prefetch(&edst[e + stride], 0, 0);
            __builtin_prefetch(&eval[e + stride], 0, 0);
        }
        bool  ok = false;
        int   sc = 0, dc = 0;
        float v  = 0.0f;
        if (e < nedges) {
            int s = esrc[e];
            int so = remap_node(s);
            if (so >= 0) {
                int d  = edst[e];
                int dofs = remap_node(d);
                if (dofs >= 0) {
                    ok = true;
                    sc = so;
                    dc = dofs;
                    v  = eval[e];
                }
            }
        }
        unsigned mask = __builtin_amdgcn_ballot_w32(ok);
        while (mask) {
            int l = __builtin_ctz(mask);
            mask &= mask - 1;
            int   bsc = __builtin_amdgcn_readlane(sc, l);
            int   bdc = __builtin_amdgcn_readlane(dc, l);
            float bv  = __int_as_float(
                            __builtin_amdgcn_readlane(__float_as_int(v), l));
            // 64 columns: lane handles c=lane and c=lane+32
            atomicAdd(&eff[bsc + lane],       bv * x[bdc + lane]);
            atomicAdd(&eff[bsc + lane + 32],  bv * x[bdc + lane + 32]);
        }
    }
}

// ---------------------------------------------------------------------------
// MLP gate + Euler update (WMMA f32 16x16x4, exact f32 precision).
// Block = 128 threads (4 waves), owns 16 rows of x. Wave w computes the
// 16x16 tile for columns [16w, 16w+16).
//   layer1: h = relu(x @ w1 + b1)      K = 128  -> 32 wmma / tile
//   layer2: g = sigmoid(h @ w2 + b2)   K = 64   -> 16 wmma / tile
//   x      = x + dt * g * (eff - x)    (both halves share gate g)
// A fragment (16x4 f32): lane m (0..15) holds M=m; K split: lanes<16 hold
//   K={kb,kb+1}, lanes>=16 hold K={kb+2,kb+3} (vgpr = K%2).
// B fragment (4x16 f32): lanes 0..15 carry N=0..15; vgpr r holds K=r for
//   lanes<16 and K=r+2 for lanes>=16 (mirrors the C-matrix row striping).
// ---------------------------------------------------------------------------
__global__ void __launch_bounds__(128)
k_mlp_update(float* __restrict__ x,
             const float* __restrict__ eff,
             const float* __restrict__ w1,
             const float* __restrict__ b1,
             const float* __restrict__ w2,
             const float* __restrict__ b2,
             const float* __restrict__ ts,
             int step) {
    __shared__ float hlds[16][68];          // padded: no LDS bank conflicts

    const int wv   = threadIdx.x >> 5;      // wave 0..3 -> N tile
    const int lane = threadIdx.x & 31;
    const int lm   = lane & 15;
    const int lh   = lane >> 4;
    const int row0 = blockIdx.x * 16;
    const int col  = wv * 16 + lm;          // output column 0..63

    // ---- layer 1: 16x64 = x[row0:row0+16, 0:128] @ w1 -------------------
    v8f acc = {0.f, 0.f, 0.f, 0.f, 0.f, 0.f, 0.f, 0.f};
    for (int kb = 0; kb < 2 * DD; kb += 4) {
        int krow = kb + 2 * lh;
        v2f a = *(const v2f*)(x + (row0 + lm) * XCOLS + krow);
        v2f b;
        b.x = w1[(krow + 0) * DD + col];
        b.y = w1[(krow + 1) * DD + col];
        acc = __builtin_amdgcn_wmma_f32_16x16x4_f32(
                  false, a, false, b, (short)0, acc, false, false);
    }
    {
        float bc = b1[col];
        #pragma unroll
        for (int r = 0; r < 8; ++r) {
            float hv = acc[r] + bc;
            hlds[r + 8 * lh][col] = hv > 0.f ? hv : 0.f;
        }
    }
    __syncthreads();

    // ---- layer 2: gate = sigmoid(h @ w2 + b2) ---------------------------
    v8f acc2 = {0.f, 0.f, 0.f, 0.f, 0.f, 0.f, 0.f, 0.f};
    for (int kb = 0; kb < DD; kb += 4) {
        int krow = kb + 2 * lh;
        v2f a = *(const v2f*)(&hlds[lm][krow]);
        v2f b;
        b.x = w2[(krow + 0) * DD + col];
        b.y = w2[(krow + 1) * DD + col];
        acc2 = __builtin_amdgcn_wmma_f32_16x16x4_f32(
                   false, a, false, b, (short)0, acc2, false, false);
    }

    // ---- fused Euler update: x += dt * g * (eff - x) --------------------
    const float b2c = b2[col];
    const float dt  = ts[step + 1] - ts[step];
    #pragma unroll
    for (int r = 0; r < 8; ++r) {
        float g   = 1.f / (1.f + __expf(-(acc2[r] + b2c)));
        int   row = row0 + r + 8 * lh;
        int   iu  = row * XCOLS + col;
        int   ii  = iu + DD;
        float xu  = x[iu];
        float xi  = x[ii];
        x[iu] = xu + dt * g * (eff[iu] - xu);
        x[ii] = xi + dt * g * (eff[ii] - xi);
    }
}

// ---------------------------------------------------------------------------
// out[b] = sigmoid( dot(x[b, 0:64], x[b, 64:128]) )    one wave per row
// ---------------------------------------------------------------------------
__global__ void k_final(const float* __restrict__ x, float* __restrict__ out) {
    int gwave = (blockIdx.x * blockDim.x + threadIdx.x) >> 5;
    int lane  = threadIdx.x & 31;
    if (gwave >= BATCH) return;
    const float* row = x + gwave * XCOLS;
    float p = row[lane] * row[DD + lane] + row[lane + 32] * row[DD + lane + 32];
    #pragma unroll
    for (int off = 16; off; off >>= 1)
        p += __shfl_xor(p, off, 32);
    if (lane == 0)
        out[gwave] = 1.f / (1.f + __expf(-p));
}

// ---------------------------------------------------------------------------
extern "C" void kernel_launch(void* const* d_in, const int* in_sizes, int n_in,
                              void* d_out, int out_size, void* d_ws, size_t ws_size,
                              hipStream_t stream) {
    const int*   users  = (const int*)  d_in[0];
    const int*   items  = (const int*)  d_in[1];
    const float* uemb   = (const float*)d_in[2];
    const float* iemb   = (const float*)d_in[3];
    const float* w1     = (const float*)d_in[4];
    const float* b1     = (const float*)d_in[5];
    const float* w2     = (const float*)d_in[6];
    const float* b2     = (const float*)d_in[7];
    const int*   esrc   = (const int*)  d_in[8];
    const int*   edst   = (const int*)  d_in[9];
    const float* evals  = (const float*)d_in[10];
    const float* tsteps = (const float*)d_in[11];
    const int    nedges = in_sizes[8];

    float* x   = (float*)d_ws;                 // 8192*128 f32 = 4 MB
    float* eff = x + BATCH * XCOLS;            // 8192*128 f32 = 4 MB
    float* out = (float*)d_out;

    k_init_x<<<(BATCH * XCOLS + 255) / 256, 256, 0, stream>>>(
        users, items, uemb, iemb, x);

    for (int s = 0; s < 3; ++s) {              // ODE_STEPS-1 Euler steps
        k_zero<<<1024, 256, 0, stream>>>(eff, BATCH * XCOLS);
        k_scatter<<<1024, 256, 0, stream>>>(esrc, edst, evals, x, eff, nedges);
        k_mlp_update<<<BATCH / 16, 128, 0, stream>>>(
            x, eff, w1, b1, w2, b2, tsteps, s);
    }

    k_final<<<(BATCH * 32 + 255) / 256, 256, 0, stream>>>(x, out);
}